// GCN_noAtt_39058432590434
// MI455X (gfx1250) — compile-verified
//
#include <hip/hip_runtime.h>

#define DFEAT 64
#define NEG_SLOPE 0.01f

typedef __attribute__((ext_vector_type(2))) float v2f;
typedef __attribute__((ext_vector_type(8))) float v8f;

// Non-returning native f32 atomic add (GLOBAL_ATOMIC_ADD_F32, STOREcnt-tracked).
__device__ __forceinline__ void gatomic_add_f32(float* addr, float val) {
  asm volatile("global_atomic_add_f32 %0, %1, off" : : "v"(addr), "v"(val) : "memory");
}

// Async global -> LDS copy of one DWORD (GLOBAL_LOAD_ASYNC_TO_LDS_B32,
// ASYNCcnt-tracked). lds_off is the wave-relative LDS byte address
// (= low 32 bits of the flat pointer, since the LDS aperture is 4GB-aligned).
__device__ __forceinline__ void async_ld_lds_b32(unsigned lds_off, const float* gaddr) {
  asm volatile("global_load_async_to_lds_b32 %0, %1, off"
               : : "v"(lds_off), "v"(gaddr) : "memory");
}

__device__ __forceinline__ void wait_asynccnt0() {
  asm volatile("s_wait_asynccnt 0x0" ::: "memory");
}

__global__ void zero_f32(float* __restrict__ p, int n) {
  int i = blockIdx.x * blockDim.x + threadIdx.x;
  int stride = gridDim.x * blockDim.x;
  for (; i < n; i += stride) p[i] = 0.0f;
}

// One wave (32 lanes) per edge: each lane moves 2 of the 64 features.
__global__ void edge_scatter(const float* __restrict__ feat,
                             const int* __restrict__ src,
                             const int* __restrict__ dst,
                             float* __restrict__ agg, int nedges) {
  int wave = (int)((blockIdx.x * blockDim.x + threadIdx.x) >> 5);
  int lane = threadIdx.x & 31;
  if (wave >= nedges) return;
  int s = src[wave];
  int d = dst[wave];
  v2f v = *(const v2f*)(feat + (long)s * DFEAT + lane * 2);
  float* base = agg + (long)d * DFEAT + lane * 2;
  gatomic_add_f32(base + 0, v.x);
  gatomic_add_f32(base + 1, v.y);
}

// out = leaky_relu(agg @ Wrel + bias + xin @ Wroot), optionally row-gathered.
// One wave computes a 16(row) x 64(col) tile via V_WMMA_F32_16X16X4_F32.
// Weights are staged once per workgroup into LDS via async global->LDS loads,
// pre-swizzled so each lane's B-fragment {W[k][n], W[k+1][n]} is a contiguous
// 8-byte pair -> one ds_load_b64 per WMMA in the K-loop.
// A layout (16x4 f32): lanes 0-15 hold K={4k,4k+1}, lanes 16-31 hold K={4k+2,4k+3}.
// B layout (4x16 f32): VGPR0 = K-row (0 | 2), VGPR1 = K-row (1 | 3), N = lane&15.
// C/D layout: VGPR i -> row m0+i (lanes 0-15) / m0+8+i (lanes 16-31), col = lane&15.
__global__ void gcn_gemm(const float* __restrict__ agg,
                         const float* __restrict__ xin,
                         const float* __restrict__ Wrel,   // [D,D] row-major
                         const float* __restrict__ Wroot,  // [D,D] row-major
                         const float* __restrict__ bias,   // [D]
                         const int* __restrict__ gather_idx,  // may be null
                         float* __restrict__ out,
                         int nrows_out) {
  // LDS: 2 matrices * 64*64 f32 pre-swizzled: idx(m,k,n) = m*4096 + (k>>1)*128 + n*2 + (k&1)
  __shared__ float wlds[2 * DFEAT * DFEAT];

  int tid = threadIdx.x;
  // Cooperative async staging of both weight matrices (all 8 waves participate).
  {
    unsigned lds_base = (unsigned)(uintptr_t)(&wlds[0]);
#pragma unroll
    for (int m = 0; m < 2; ++m) {
      const float* W = m ? Wroot : Wrel;
#pragma unroll
      for (int i = 0; i < 16; ++i) {
        int elem = i * 256 + tid;          // 0..4095
        int k = elem >> 6;
        int n = elem & 63;
        unsigned idx = (unsigned)(m * 4096 + ((k >> 1) * 128) + n * 2 + (k & 1));
        async_ld_lds_b32(lds_base + idx * 4u, W + elem);
      }
    }
    wait_asynccnt0();
  }
  __syncthreads();

  int wave = tid >> 5;
  int lane = tid & 31;
  int m0 = ((int)blockIdx.x * 8 + wave) * 16;
  if (m0 < nrows_out) {  // wave-uniform predicate: EXEC stays all-1s inside
    int lrow = lane & 15;
    int khalf = (lane >> 4) * 2;  // 0 or 2
    int r = m0 + lrow;
    int rc = r < nrows_out ? r : nrows_out - 1;
    long arow = gather_idx ? (long)gather_idx[rc] : (long)rc;

    // Preload A fragments for all 16 K-steps, both input matrices.
    v2f a1[16], a2[16];
    const float* pa1 = agg + arow * DFEAT + khalf;
    const float* pa2 = xin + arow * DFEAT + khalf;
#pragma unroll
    for (int kk = 0; kk < 16; ++kk) {
      a1[kk] = *(const v2f*)(pa1 + kk * 4);
      a2[kk] = *(const v2f*)(pa2 + kk * 4);
    }

#pragma unroll
    for (int nt = 0; nt < 4; ++nt) {
      int col = nt * 16 + lrow;
      float bv = bias[col];
      v8f acc = {bv, bv, bv, bv, bv, bv, bv, bv};
#pragma unroll
      for (int kk = 0; kk < 16; ++kk) {
        int kb = kk * 4 + khalf;  // even
        int pidx = (kb >> 1) * 128 + col * 2;
        v2f brel = *(const v2f*)(&wlds[pidx]);            // {Wrel[kb][col], Wrel[kb+1][col]}
        acc = __builtin_amdgcn_wmma_f32_16x16x4_f32(false, a1[kk], false, brel,
                                                    (short)0, acc, false, false);
        v2f broot = *(const v2f*)(&wlds[4096 + pidx]);    // {Wroot[kb][col], Wroot[kb+1][col]}
        acc = __builtin_amdgcn_wmma_f32_16x16x4_f32(false, a2[kk], false, broot,
                                                    (short)0, acc, false, false);
      }
      // Fused leaky ReLU.
#pragma unroll
      for (int i = 0; i < 8; ++i) {
        float v = acc[i];
        acc[i] = v > 0.0f ? v : v * NEG_SLOPE;
      }
      int mbase = m0 + (lane >> 4) * 8;
#pragma unroll
      for (int i = 0; i < 8; ++i) {
        int mrow = mbase + i;
        if (mrow < nrows_out) out[(long)mrow * DFEAT + col] = acc[i];
      }
    }
  }
}

extern "C" void kernel_launch(void* const* d_in, const int* in_sizes, int n_in,
                              void* d_out, int out_size, void* d_ws, size_t ws_size,
                              hipStream_t stream) {
  const float* x       = (const float*)d_in[0];
  const int*   ei      = (const int*)d_in[1];
  const int*   bbox    = (const int*)d_in[2];
  const float* W1_rel  = (const float*)d_in[3];
  const float* b1      = (const float*)d_in[4];
  const float* W1_root = (const float*)d_in[5];
  const float* W2_rel  = (const float*)d_in[6];
  const float* b2      = (const float*)d_in[7];
  const float* W2_root = (const float*)d_in[8];

  int nnodes = in_sizes[0] / DFEAT;
  int nedges = in_sizes[1] / 2;
  int nbbox  = in_sizes[2];
  const int* src = ei;
  const int* dst = ei + nedges;

  float* agg = (float*)d_ws;                      // [nnodes, D]
  float* h   = agg + (size_t)nnodes * DFEAT;      // [nnodes, D]
  float* outp = (float*)d_out;                    // [nbbox, D]

  int cnt = nnodes * DFEAT;
  int scatter_blocks = (nedges + 7) / 8;          // 8 waves (edges) per 256-thr block
  int tiles1 = (nnodes + 15) / 16;
  int tiles2 = (nbbox + 15) / 16;

  // Layer 1
  zero_f32<<<2048, 256, 0, stream>>>(agg, cnt);
  edge_scatter<<<scatter_blocks, 256, 0, stream>>>(x, src, dst, agg, nedges);
  gcn_gemm<<<(tiles1 + 7) / 8, 256, 0, stream>>>(agg, x, W1_rel, W1_root, b1,
                                                 nullptr, h, nnodes);
  // Layer 2 (only bbox rows materialized through the GEMM)
  zero_f32<<<2048, 256, 0, stream>>>(agg, cnt);
  edge_scatter<<<scatter_blocks, 256, 0, stream>>>(h, src, dst, agg, nedges);
  gcn_gemm<<<(tiles2 + 7) / 8, 256, 0, stream>>>(agg, h, W2_rel, W2_root, b2,
                                                 bbox, outp, nbbox);
}